// MeshNN_26680336843569
// MI455X (gfx1250) — compile-verified
//
#include <hip/hip_runtime.h>

// MeshNN hat-basis GEMV, fused, f32 WMMA (V_WMMA_F32_16X16X4_F32), gfx1250.
//
// u[p] = sum_{j=1..510} hat_j(x[p]) * v[j-1]  +  w0*phi_0(x[p]) + w1*phi_L(x[p])
//
// One wave32 owns 16 points (M=16). K-loop over interior nodes (510, padded
// to 512) in steps of 4 using V_WMMA_F32_16X16X4_F32. Weight v is folded into
// the A matrix (a = phi * w); B is the all-ones matrix so every column of D
// is the GEMV result and the B VGPR layout cannot affect correctness.
// A layout (ISA 7.12.2, 32-bit A 16x4): lane L<16 -> row M=L, K={k0,k0+1};
// lane L>=16 -> row M=L-16, K={k0+2,k0+3}.
// Padded K slots use invL=invR=0, w=0 => phi evaluates to finite 1.0 and
// a = 1*0 = 0 exactly (no Inf/NaN, no branch, EXEC stays all-ones for WMMA).

typedef __attribute__((ext_vector_type(2))) float v2f;
typedef __attribute__((ext_vector_type(8))) float v8f;

#define N_NODES_C 512
#define KDIM 512                       // interior nodes (510) padded to mult of 4
#define GHOST_EPS_F 0.001f
#define WAVES_PER_BLOCK 8
#define POINTS_PER_WAVE 16
#define POINTS_PER_BLOCK (WAVES_PER_BLOCK * POINTS_PER_WAVE)   // 128

__global__ __launch_bounds__(256)
void meshnn_hat_gemv_wmma(const float* __restrict__ x,
                          const float* __restrict__ coords,
                          const float* __restrict__ nodal,
                          const float* __restrict__ dd,
                          float* __restrict__ out,
                          int npoints, int nnodes)
{
    // Per-node packed constants: {c_j, 1/(c_j-c_{j-1}), 1/(c_{j+1}-c_j), v_{j-1}}
    __shared__ float4 node[KDIM];
    __shared__ float  uu[WAVES_PER_BLOCK][POINTS_PER_WAVE];

    const int tid = threadIdx.x;

    // Cooperative build of node table (j = k+1, k = 0..nnodes-3 valid).
    for (int k = tid; k < KDIM; k += blockDim.x) {
        float cm = 0.0f, il = 0.0f, ir = 0.0f, w = 0.0f;
        if (k < nnodes - 2) {
            const float cl = coords[k];
            cm = coords[k + 1];
            const float cr = coords[k + 2];
            il = 1.0f / (cm - cl);
            ir = 1.0f / (cr - cm);
            w  = nodal[k];
        }
        node[k] = make_float4(cm, il, ir, w);
    }
    __syncthreads();

    const int lane  = tid & 31;
    const int wave  = tid >> 5;
    const int tile0 = blockIdx.x * POINTS_PER_BLOCK + wave * POINTS_PER_WAVE;
    const int m     = lane & 15;               // matrix row owned by this lane
    const int pidx  = tile0 + m;
    const float xm  = (pidx < npoints) ? x[pidx] : 0.0f;
    const int koff  = (lane >> 4) << 1;        // 0 for lanes 0-15, 2 for 16-31

    const v2f ones = {1.0f, 1.0f};             // B = all-ones (layout-agnostic)
    v8f acc = {};                               // C/D accumulator, 16x16 f32

    #pragma unroll 4
    for (int k0 = 0; k0 < KDIM; k0 += 4) {
        const int kk = k0 + koff;
        const float4 n0 = node[kk];            // ds_load_b128, broadcast
        const float4 n1 = node[kk + 1];

        // hat(x; c_{j-1}, c_j, c_{j+1}) = relu(1 - relu(c_j-x)*invL)
        //                               + relu(1 - relu(x-c_j)*invR) - 1
        float t0l = fmaxf(n0.x - xm, 0.0f);
        float t0r = fmaxf(xm - n0.x, 0.0f);
        float phi0 = fmaxf(1.0f - t0l * n0.y, 0.0f)
                   + fmaxf(1.0f - t0r * n0.z, 0.0f) - 1.0f;
        float t1l = fmaxf(n1.x - xm, 0.0f);
        float t1r = fmaxf(xm - n1.x, 0.0f);
        float phi1 = fmaxf(1.0f - t1l * n1.y, 0.0f)
                   + fmaxf(1.0f - t1r * n1.z, 0.0f) - 1.0f;

        v2f a;
        a.x = phi0 * n0.w;                     // fold nodal weight into A
        a.y = phi1 * n1.w;

        // D = A(16x4) * ones(4x16) + C   -> every column n holds u_uu partials
        acc = __builtin_amdgcn_wmma_f32_16x16x4_f32(
            /*neg_a=*/false, a, /*neg_b=*/false, ones,
            /*c_mod=*/(short)0, acc, /*reuse_a=*/false, /*reuse_b=*/false);
    }

    // Column n=0 lives in lanes 0 (rows 0-7) and 16 (rows 8-15), VGPRs 0..7.
    if ((lane & 15) == 0) {
        const int rbase = (lane >> 4) * 8;
        #pragma unroll
        for (int r = 0; r < 8; ++r) uu[wave][rbase + r] = acc[r];
    }
    __syncthreads();

    // Epilogue: rank-2 boundary term + store (lanes 0-15 of each wave).
    if (lane < 16 && pidx < npoints) {
        const float cA0 = coords[0];
        const float cA1 = coords[1];
        const float cB0 = coords[nnodes - 2];
        const float cB1 = coords[nnodes - 1];
        const float w0 = dd[0], w1 = dd[1];

        // phi_0 = hat(x; cA0-eps, cA0, cA1)
        float l = fmaxf(1.0f - fmaxf(cA0 - xm, 0.0f) / GHOST_EPS_F, 0.0f);
        float r = fmaxf(1.0f - fmaxf(xm - cA0, 0.0f) / (cA1 - cA0), 0.0f);
        const float phi_0 = l + r - 1.0f;
        // phi_L = hat(x; cB0, cB1, cB1+eps)
        l = fmaxf(1.0f - fmaxf(cB1 - xm, 0.0f) / (cB1 - cB0), 0.0f);
        r = fmaxf(1.0f - fmaxf(xm - cB1, 0.0f) / GHOST_EPS_F, 0.0f);
        const float phi_L = l + r - 1.0f;

        out[pidx] = uu[wave][m] + w0 * phi_0 + w1 * phi_L;
    }
}

extern "C" void kernel_launch(void* const* d_in, const int* in_sizes, int n_in,
                              void* d_out, int out_size, void* d_ws, size_t ws_size,
                              hipStream_t stream) {
    const float* x      = (const float*)d_in[0];   // (N_POINTS,1) f32
    const float* coords = (const float*)d_in[1];   // (512,) f32
    const float* nodal  = (const float*)d_in[2];   // (510,) f32
    const float* dd     = (const float*)d_in[3];   // (2,) f32
    float* out = (float*)d_out;                    // (N_POINTS,1) f32

    const int npoints = out_size;
    const int nnodes  = in_sizes[1];               // 512
    const int blocks  = (npoints + POINTS_PER_BLOCK - 1) / POINTS_PER_BLOCK;

    meshnn_hat_gemv_wmma<<<blocks, WAVES_PER_BLOCK * 32, 0, stream>>>(
        x, coords, nodal, dd, out, npoints, nnodes);
}